// ScriptEGRUD_72705206387128
// MI455X (gfx1250) — compile-verified
//
#include <hip/hip_runtime.h>
#include <hip/hip_bf16.h>
#include <math.h>

// ---------------------------------------------------------------------------
// ScriptEGRU forward on MI455X (gfx1250, wave32).
//   Phase 1: xW = x @ W^T + b_w (fp32 WMMA 16x16x4, 4 N-tiles/wave sharing the
//            A fragment), written into the I_s output region ([T,B,3H]).
//   Phase 2: 256 serial fused step kernels: hU GEMM (fp32 WMMA, A fragments
//            staged to LDS by the Tensor Data Mover with double buffering and
//            s_wait_tensorcnt) + gates + spike + state update.
//            U (48MB) stays L2-resident (192MB L2).
// ---------------------------------------------------------------------------

#define B_   64
#define T_   256
#define I_   1024
#define H_   2048
#define N3H_ (3 * H_)

typedef __attribute__((ext_vector_type(2))) float v2f;
typedef __attribute__((ext_vector_type(8))) float v8f;

#if defined(__HIP_DEVICE_COMPILE__) && __has_builtin(__builtin_amdgcn_tensor_load_to_lds)
#define HAVE_TDM 1
#else
#define HAVE_TDM 0
#endif

__device__ __forceinline__ float sigm(float x) { return 1.0f / (1.0f + __expf(-x)); }

// --------------------------- init: zero states + tr -------------------------
__global__ void init_kernel(float* __restrict__ hA, float* __restrict__ hB,
                            float* __restrict__ cst, float* __restrict__ tr,
                            size_t trN) {
    size_t i = (size_t)blockIdx.x * blockDim.x + threadIdx.x;
    size_t stride = (size_t)gridDim.x * blockDim.x;
    for (size_t j = i; j < (size_t)B_ * H_; j += stride) {
        hA[j] = 0.0f; hB[j] = 0.0f; cst[j] = 0.0f;
    }
    for (size_t j = i; j < trN; j += stride) tr[j] = 0.0f;
}

// ------------------- Phase 1: xW = x @ W^T + b_w  ([T*B, 3H]) ---------------
// 8 waves/block; each wave owns a 16x64 strip: one A fragment feeds 4 WMMAs.
// fp32 WMMA A layout (16x4): lanes 0-15 -> K=0,1 ; lanes 16-31 -> K=2,3.
__global__ __launch_bounds__(256) void xw_gemm_kernel(
    const float* __restrict__ x,   // [B, T, I]
    const float* __restrict__ W,   // [3H, I]
    const float* __restrict__ bw,  // [3H]
    float* __restrict__ out)       // [T*B, 3H]  (I_s region of d_out)
{
    const int lane = threadIdx.x & 31;
    const int wave = threadIdx.x >> 5;
    const int NG = (N3H_ / 16) / 4;                 // 96 strip-groups along N
    const int gid = blockIdx.x * 8 + wave;
    const int m0 = (gid / NG) * 16;
    const int n0 = (gid % NG) * 64;

    // A fragment source: output row m = t*B + b  -> x row = b*T + t
    const int mrow = m0 + (lane & 15);
    const int tt = mrow >> 6;                       // / B_ (=64)
    const int bb = mrow & 63;
    const int khalf = (lane >> 4) << 1;             // 0 or 2
    const float* ap = x + (size_t)(bb * T_ + tt) * I_ + khalf;

    // B fragment sources: B[k][n] = W[n][k]
    const float* bp[4];
    int ncol[4];
#pragma unroll
    for (int j = 0; j < 4; ++j) {
        ncol[j] = n0 + j * 16 + (lane & 15);
        bp[j] = W + (size_t)ncol[j] * I_ + khalf;
    }

    v8f acc[4] = {};
#pragma unroll 4
    for (int k = 0; k < I_; k += 4) {
        __builtin_prefetch(ap + k + 256, 0, 0);     // global_prefetch ahead of A
        v2f af = *(const v2f*)(ap + k);
#pragma unroll
        for (int j = 0; j < 4; ++j) {
            v2f bf = *(const v2f*)(bp[j] + k);
            acc[j] = __builtin_amdgcn_wmma_f32_16x16x4_f32(
                false, af, false, bf, (short)0, acc[j], false, false);
        }
    }

    const int rbase = m0 + ((lane >> 4) << 3);      // lanes 16-31 hold M=8..15
#pragma unroll
    for (int j = 0; j < 4; ++j) {
        const float bias = bw[ncol[j]];
#pragma unroll
        for (int r = 0; r < 8; ++r)
            out[(size_t)(rbase + r) * N3H_ + ncol[j]] = acc[j][r] + bias;
    }
}

// ----------------- TDM: stage one 64x64 fp32 chunk of h into LDS ------------
#if HAVE_TDM
typedef __attribute__((ext_vector_type(4))) unsigned int v4u;
typedef __attribute__((ext_vector_type(4))) int v4i;
typedef __attribute__((ext_vector_type(8))) int v8i;

// 2D tile: tile_dim0 = 64 elems (row chunk), tile_dim1 = 64 rows,
// tensor_dim0_stride = H. TDM pad: +2 DWORDs every 64 DWORDs -> LDS row
// stride of 66 floats (8B-aligned rows, bank-conflict-free ds_load_b64).
__device__ __forceinline__ void tdm_load_h_chunk(const float* gsrc,
                                                 unsigned lds_byte_off) {
    unsigned long long ga = (unsigned long long)gsrc;
    v4u g0;
    g0[0] = 1u;                                         // count=1 (valid D#)
    g0[1] = lds_byte_off;                               // lds_addr
    g0[2] = (unsigned)(ga & 0xffffffffu);               // global_addr[31:0]
    g0[3] = (unsigned)((ga >> 32) & 0x01ffffffu) | (2u << 30); // addr hi | type=2
    v8i g1;
    g1[0] = (int)((2u << 16) |                          // data_size = 4B
                  (1u << 20) |                          // pad_enable
                  (5u << 22) |                          // pad_interval = 64 DW
                  (1u << 25));                          // pad_amount = 2 DW
    g1[1] = (int)(((unsigned)H_ & 0xffffu) << 16);      // tensor_dim0 lo16
    g1[2] = (int)((64u << 16) | ((unsigned)H_ >> 16));  // tensor_dim1=64 | dim0 hi
    g1[3] = (int)(64u << 16);                           // tile_dim0 = 64
    g1[4] = 64;                                         // tile_dim1 = 64
    g1[5] = H_;                                         // tensor_dim0_stride lo32
    g1[6] = 0;
    g1[7] = 0;
    v4i z4 = {0, 0, 0, 0};
#if __clang_major__ >= 23
    v8i z8 = {0, 0, 0, 0, 0, 0, 0, 0};
    __builtin_amdgcn_tensor_load_to_lds(g0, g1, z4, z4, z8, 0);
#else
    __builtin_amdgcn_tensor_load_to_lds(g0, g1, z4, z4, 0);
#endif
}
#endif

// ------------- Phase 2: one recurrent step, fully fused ---------------------
// Grid = H/64 = 32 blocks. Block owns 64 H-columns: computes h_u/h_r/h_c
// (48 16x16 WMMA tiles, 6 per wave with a fixed M-tile so one LDS A fragment
// feeds 6 WMMAs), then the elementwise GRU/spike epilogue.
__global__ __launch_bounds__(256) void step_kernel(
    const float* __restrict__ U,        // [3H, H]
    const float* __restrict__ bu,       // [3H]
    const float* __restrict__ thr_rep,  // [H]
    float* __restrict__ Iio,            // [T*B, 3H]: xW in, i_* out (in place)
    float* __restrict__ hs,             // [B, T, H]
    float* __restrict__ cout,           // [T, B, H]
    float* __restrict__ oout,           // [T, B, H]
    const float* __restrict__ h_in,     // [B, H]
    float* __restrict__ h_out,          // [B, H]
    float* __restrict__ c_state,        // [B, H]
    int t)
{
    // ht (TDM staging, double-buffered) and gates are time-disjoint -> union.
    __shared__ union {
        float ht[2][64][66];            // 66 = 64 + TDM pad (2 DW): row = 264B
        float gates[3][64][65];
    } sm;

    const int lane = threadIdx.x & 31;
    const int wave = threadIdx.x >> 5;
    const int hcol0 = blockIdx.x * 64;

    const int mt = wave & 3;            // fixed M-tile -> one A frag per k-step
    const int cg = wave >> 2;           // column group: 6 of 12 (gate,nt) pairs
    const int khalf = (lane >> 4) << 1;
    const int arow = mt * 16 + (lane & 15);

    const float* bp[6];
    int gcol[6];
#pragma unroll
    for (int i = 0; i < 6; ++i) {
        const int p = cg * 6 + i;       // 0..11
        const int g = p >> 2;           // gate 0..2
        const int nt = p & 3;           // n-tile 0..3
        gcol[i] = g * H_ + hcol0 + nt * 16 + (lane & 15);
        bp[i] = U + (size_t)gcol[i] * H_ + khalf;
    }

    v8f acc[6] = {};
    const int KC = 64;
    const int NCH = H_ / KC;            // 32 K-chunks

#if HAVE_TDM
    if (wave == 0)                      // EXEC-independent DMA; one issue/WG
        tdm_load_h_chunk(h_in, (unsigned)(unsigned long long)&sm.ht[0][0][0]);
#endif
    for (int c = 0; c < NCH; ++c) {
#if HAVE_TDM
        const int cur = c & 1;
        if (wave == 0) {
            if (c + 1 < NCH) {          // prefetch next chunk into other buffer
                tdm_load_h_chunk(h_in + (size_t)(c + 1) * KC,
                                 (unsigned)(unsigned long long)&sm.ht[cur ^ 1][0][0]);
                // TDM completes in order: <=1 outstanding => chunk c landed.
                __builtin_amdgcn_s_wait_tensorcnt(1);
            } else {
                __builtin_amdgcn_s_wait_tensorcnt(0);
            }
        }
        __syncthreads();
#else
        const int cur = 0;
        __syncthreads();
        for (int j = threadIdx.x; j < 64 * KC; j += 256)
            sm.ht[0][j >> 6][j & 63] =
                h_in[(size_t)(j >> 6) * H_ + (size_t)c * KC + (j & 63)];
        __syncthreads();
#endif
        const int kg = c * KC;
#pragma unroll 4
        for (int kl = 0; kl < KC; kl += 4) {
            v2f af = *(const v2f*)&sm.ht[cur][arow][kl + khalf];
#pragma unroll
            for (int i = 0; i < 6; ++i) {
                v2f bf = *(const v2f*)(bp[i] + kg + kl);
                acc[i] = __builtin_amdgcn_wmma_f32_16x16x4_f32(
                    false, af, false, bf, (short)0, acc[i], false, false);
            }
        }
        __syncthreads();                // protect staging buffer before reuse
    }

    // Park gate tiles (+ b_u) in LDS for the fused epilogue.
    const int rbase = mt * 16 + ((lane >> 4) << 3);
#pragma unroll
    for (int i = 0; i < 6; ++i) {
        const int p = cg * 6 + i;
        const int g = p >> 2;
        const int nt = p & 3;
        const float bias = bu[gcol[i]];
        const int cc = nt * 16 + (lane & 15);
#pragma unroll
        for (int r = 0; r < 8; ++r)
            sm.gates[g][rbase + r][cc] = acc[i][r] + bias;
    }
    __syncthreads();

    // Fused GRU + spike epilogue over the 64x64 (batch x hcol) slab.
    for (int j = threadIdx.x; j < 64 * 64; j += 256) {
        const int b = j >> 6;
        const int cc = j & 63;
        const int hc = hcol0 + cc;
        const size_t rI = ((size_t)t * B_ + b) * N3H_;

        const float xu = Iio[rI + hc];
        const float xr = Iio[rI + H_ + hc];
        const float xc = Iio[rI + 2 * H_ + hc];

        // ALPHA_G = 0: i_* = x_* + h_* (no dependence on previous I state)
        const float iu = xu + sm.gates[0][b][cc];
        const float ir = xr + sm.gates[1][b][cc];
        const float u  = sigm(iu);
        const float r  = sigm(ir);
        const float ic = xc + r * sm.gates[2][b][cc];
        const float z  = tanhf(ic);

        const float cprev = c_state[(size_t)b * H_ + hc];
        const float cnew  = u * cprev + (1.0f - u) * z;
        const float thv   = sigm(thr_rep[hc]);
        const float o     = (cnew - thv) > 0.0f ? 1.0f : 0.0f;
        const float hn    = cnew * o;
        const float cres  = cnew - o * thv;

        Iio[rI + hc]          = iu;
        Iio[rI + H_ + hc]     = ir;
        Iio[rI + 2 * H_ + hc] = ic;

        hs[((size_t)b * T_ + t) * H_ + hc] = hn;
        const size_t rO = ((size_t)t * B_ + b) * H_ + hc;
        cout[rO] = (t == T_ - 1) ? cnew : cres;   // un-reset c at final step
        oout[rO] = o;

        c_state[(size_t)b * H_ + hc] = cres;      // column-owned: in place
        h_out[(size_t)b * H_ + hc]   = hn;        // ping-pong buffer
    }
}

// ---------------------------------------------------------------------------
extern "C" void kernel_launch(void* const* d_in, const int* in_sizes, int n_in,
                              void* d_out, int out_size, void* d_ws, size_t ws_size,
                              hipStream_t stream) {
    const float* x   = (const float*)d_in[0];
    const float* W   = (const float*)d_in[1];
    const float* U   = (const float*)d_in[2];
    const float* bw  = (const float*)d_in[3];
    const float* bu  = (const float*)d_in[4];
    const float* thr = (const float*)d_in[5];

    float* out    = (float*)d_out;
    float* hs     = out;                                       // [B,T,H]
    float* c_out  = hs + (size_t)B_ * T_ * H_;                 // [T,B,H]
    float* o_out  = c_out + (size_t)T_ * B_ * H_;              // [T,B,H]
    float* I_out  = o_out + (size_t)T_ * B_ * H_;              // [T,B,3H]
    float* tr_out = I_out + (size_t)T_ * B_ * N3H_;            // [T,B,H]

    float* hA      = (float*)d_ws;                             // [B,H]
    float* hB      = hA + (size_t)B_ * H_;                     // [B,H]
    float* c_state = hB + (size_t)B_ * H_;                     // [B,H]

    init_kernel<<<2048, 256, 0, stream>>>(hA, hB, c_state, tr_out,
                                          (size_t)T_ * B_ * H_);

    // Phase 1: 1024 M-tiles x 96 N-strips, 8 strips (waves) per block.
    const int groups = (T_ * B_ / 16) * ((N3H_ / 16) / 4);     // 98304
    xw_gemm_kernel<<<groups / 8, 256, 0, stream>>>(x, W, bw, I_out);

    // Phase 2: serial scan, h double-buffered (h_0 = 0 in hA).
    float* bufs[2] = {hA, hB};
    for (int t = 0; t < T_; ++t) {
        step_kernel<<<H_ / 64, 256, 0, stream>>>(
            U, bu, thr, I_out, hs, c_out, o_out,
            bufs[t & 1], bufs[(t + 1) & 1], c_state, t);
    }
}